// RNet_33088428048694
// MI455X (gfx1250) — compile-verified
//
#include <hip/hip_runtime.h>
#include <hip/hip_bf16.h>

typedef __attribute__((ext_vector_type(2))) float v2f;
typedef __attribute__((ext_vector_type(8))) float v8f;

// ---------------- workspace layout (float offsets) ----------------
#define WS_MV_IN   0       // 2053  m_rnn input  {x0,x1,x8,x9,x10,h1,h2}
#define WS_K1_IN   2560    // 2049  {x3, h1, f1}
#define WS_K2_IN   5120    // 2049  {x6, h2, f2}
#define WS_F1_IN   7680    // 1025  {x4, k1}
#define WS_F2_IN   10240   // 1025  {x7, k2}
#define WS_H1_IN   12800   // 2049  {x2, out_m[0:1024], k1}
#define WS_H2_IN   15360   // 2049  {x5, out_m[1024:2048], k2}
#define WS_M_GI    18432   // 6144
#define WS_M_GH    24576   // 6144
#define WS_M_S     30720   // 2048
#define WS_OUT_M   32768   // 2048
#define WS_G6_GI   34816   // 6*3072
#define WS_G6_GH   53248   // 6*3072
#define WS_S6      71680   // 6*1024  (h1,h2,k1,k2,f1,f2)

__device__ __forceinline__ float sigmoidf_(float v) {
    return 1.0f / (1.0f + expf(-v));
}

// ---------------- WMMA GEMV: out[j] = dot(W[j,:], x) + bias[j] ----------------
// One wave handles 16 consecutive rows via V_WMMA_F32_16X16X4_F32.
// A tile = W[j0:j0+16, k0:k0+4]  (ISA layout: lanes 0-15 = rows, VGPR0/1 hold
// K={0,1} for lanes<16 and K={2,3} for lanes>=16).
// B = x[k0:k0+4] broadcast across all 16 columns -> every column of D equals
// the 16 row-partials; lane 0 holds rows 0-7 in c[0..7], lane 16 rows 8-15.
struct GemvJob { const float* W; const float* x; const float* b; float* out; int K; };
struct GemvBatch { GemvJob j[12]; };

__global__ __launch_bounds__(128) void gemv16_wmma_kernel(GemvBatch batch) {
    const GemvJob job = batch.j[blockIdx.y];
    const int K    = job.K;
    const int lane = threadIdx.x & 31;
    const int wv   = threadIdx.x >> 5;
    const int j0   = (blockIdx.x * 4 + wv) * 16;
    const int ln   = lane & 15;
    const int hf   = lane >> 4;                 // lane-half selects K pair
    const float* __restrict__ wrow = job.W + (size_t)(j0 + ln) * (size_t)K;
    const float* __restrict__ x    = job.x;

    v8f c = {};
    const int Km = K & ~3;
    int k0 = 0;
    for (; k0 < Km; k0 += 4) {
        const int ka = k0 + 2 * hf;
        __builtin_prefetch(wrow + ka + 256, 0, 1);   // global_prefetch_b8, stream ahead
        v2f a, b;
        a.x = wrow[ka];
        a.y = wrow[ka + 1];
        b.x = x[ka];
        b.y = x[ka + 1];
        c = __builtin_amdgcn_wmma_f32_16x16x4_f32(false, a, false, b,
                                                  (short)0, c, false, false);
    }
    if (Km < K) {                                   // zero-padded tail chunk
        const int ka = k0 + 2 * hf;
        v2f a = {0.0f, 0.0f}, b = {0.0f, 0.0f};
        if (ka < K)     { a.x = wrow[ka];     b.x = x[ka];     }
        if (ka + 1 < K) { a.y = wrow[ka + 1]; b.y = x[ka + 1]; }
        c = __builtin_amdgcn_wmma_f32_16x16x4_f32(false, a, false, b,
                                                  (short)0, c, false, false);
    }
    if (lane == 0) {
        #pragma unroll
        for (int m = 0; m < 8; ++m) job.out[j0 + m] = c[m] + job.b[j0 + m];
    } else if (lane == 16) {
        #pragma unroll
        for (int m = 0; m < 8; ++m) job.out[j0 + 8 + m] = c[m] + job.b[j0 + 8 + m];
    }
}

// ---------------- input-vector assembly ----------------
__global__ void prep1_kernel(const float* __restrict__ x,
                             const float* __restrict__ h1, const float* __restrict__ h2,
                             const float* __restrict__ k1, const float* __restrict__ k2,
                             const float* __restrict__ f1, const float* __restrict__ f2,
                             float* __restrict__ ws) {
    const int i = blockIdx.x * 256 + threadIdx.x;   // 0..1023
    float* mv  = ws + WS_MV_IN;
    float* k1i = ws + WS_K1_IN;
    float* k2i = ws + WS_K2_IN;
    float* f1i = ws + WS_F1_IN;
    float* f2i = ws + WS_F2_IN;
    if (i == 0) {
        mv[0] = x[0]; mv[1] = x[1]; mv[2] = x[8]; mv[3] = x[9]; mv[4] = x[10];
        k1i[0] = x[3]; k2i[0] = x[6]; f1i[0] = x[4]; f2i[0] = x[7];
    }
    mv[5 + i]        = h1[i];
    mv[5 + 1024 + i] = h2[i];
    k1i[1 + i]    = h1[i]; k1i[1025 + i] = f1[i];
    k2i[1 + i]    = h2[i]; k2i[1025 + i] = f2[i];
    f1i[1 + i]    = k1[i];
    f2i[1 + i]    = k2[i];
}

__global__ void prep2_kernel(const float* __restrict__ x,
                             const float* __restrict__ out_m,
                             const float* __restrict__ k1, const float* __restrict__ k2,
                             float* __restrict__ ws) {
    const int i = blockIdx.x * 256 + threadIdx.x;   // 0..1023
    float* h1i = ws + WS_H1_IN;
    float* h2i = ws + WS_H2_IN;
    if (i == 0) { h1i[0] = x[2]; h2i[0] = x[5]; }
    h1i[1 + i] = out_m[i];        h1i[1025 + i] = k1[i];
    h2i[1 + i] = out_m[1024 + i]; h2i[1025 + i] = k2[i];
}

// ---------------- GRU gate nonlinearity ----------------
__global__ void gru_gate_kernel(const float* __restrict__ gi, const float* __restrict__ gh,
                                const float* __restrict__ h, float* __restrict__ hs, int hid) {
    const int j = blockIdx.x * 256 + threadIdx.x;
    const float r = sigmoidf_(gi[j] + gh[j]);
    const float z = sigmoidf_(gi[hid + j] + gh[hid + j]);
    const float n = tanhf(gi[2 * hid + j] + r * gh[2 * hid + j]);
    hs[j] = (1.0f - z) * n + z * h[j];
}

struct Gru6States { const float* st[6]; };
__global__ void gru_gate6_kernel(const float* __restrict__ gi_base,
                                 const float* __restrict__ gh_base,
                                 Gru6States g, float* __restrict__ s6) {
    const int gy = blockIdx.y;                      // which GRU: h1,h2,k1,k2,f1,f2
    const int j  = blockIdx.x * 256 + threadIdx.x;  // 0..1023
    const float* gi = gi_base + gy * 3072;
    const float* gh = gh_base + gy * 3072;
    const float* h  = g.st[gy];
    const float r = sigmoidf_(gi[j] + gh[j]);
    const float z = sigmoidf_(gi[1024 + j] + gh[1024 + j]);
    const float n = tanhf(gi[2048 + j] + r * gh[2048 + j]);
    s6[gy * 1024 + j] = (1.0f - z) * n + z * h[j];
}

// ---------------- output heads: 6 independent 1x1024 dots ----------------
struct Heads { const float* W[6]; const float* b[6]; const float* s[6]; };
__global__ void out_heads_kernel(Heads h, float* __restrict__ out) {
    __shared__ float red[256];
    const int bidx = blockIdx.x;
    const float* W = h.W[bidx];
    const float* s = h.s[bidx];
    float acc = 0.0f;
    for (int i = threadIdx.x; i < 1024; i += 256) acc += W[i] * s[i];
    red[threadIdx.x] = acc;
    __syncthreads();
    for (int off = 128; off > 0; off >>= 1) {
        if (threadIdx.x < off) red[threadIdx.x] += red[threadIdx.x + off];
        __syncthreads();
    }
    if (threadIdx.x == 0) out[bidx] = red[0] + h.b[bidx][0];
}

// ---------------- launcher ----------------
extern "C" void kernel_launch(void* const* d_in, const int* in_sizes, int n_in,
                              void* d_out, int out_size, void* d_ws, size_t ws_size,
                              hipStream_t stream) {
    const float* x     = (const float*)d_in[0];
    const float* st_m  = (const float*)d_in[1];
    const float* st_h1 = (const float*)d_in[2];
    const float* st_h2 = (const float*)d_in[3];
    const float* st_k1 = (const float*)d_in[4];
    const float* st_k2 = (const float*)d_in[5];
    const float* st_f1 = (const float*)d_in[6];
    const float* st_f2 = (const float*)d_in[7];
    float* ws = (float*)d_ws;

    // phase 1: assemble concatenated input vectors
    prep1_kernel<<<4, 256, 0, stream>>>(x, st_h1, st_h2, st_k1, st_k2, st_f1, st_f2, ws);

    // phase 2: m_rnn gi + gh (independent -> one batched launch, R=6144 each)
    {
        GemvBatch b = {};
        b.j[0] = { (const float*)d_in[8], ws + WS_MV_IN, (const float*)d_in[10], ws + WS_M_GI, 2053 };
        b.j[1] = { (const float*)d_in[9], st_m,          (const float*)d_in[11], ws + WS_M_GH, 2048 };
        gemv16_wmma_kernel<<<dim3(96, 2), 128, 0, stream>>>(b);
    }
    gru_gate_kernel<<<8, 256, 0, stream>>>(ws + WS_M_GI, ws + WS_M_GH, st_m, ws + WS_M_S, 2048);

    // phase 3: m_out (2048x2048)
    {
        GemvBatch b = {};
        b.j[0] = { (const float*)d_in[12], ws + WS_M_S, (const float*)d_in[13], ws + WS_OUT_M, 2048 };
        gemv16_wmma_kernel<<<dim3(32, 1), 128, 0, stream>>>(b);
    }
    prep2_kernel<<<4, 256, 0, stream>>>(x, ws + WS_OUT_M, st_k1, st_k2, ws);

    // phase 4: all six small GRUs' gi/gh in one batched launch (R=3072 each)
    {
        GemvBatch b = {};
        // h_rnn1
        b.j[0]  = { (const float*)d_in[14], ws + WS_H1_IN, (const float*)d_in[16], ws + WS_G6_GI + 0,     2049 };
        b.j[1]  = { (const float*)d_in[15], st_h1,         (const float*)d_in[17], ws + WS_G6_GH + 0,     1024 };
        // h_rnn2
        b.j[2]  = { (const float*)d_in[18], ws + WS_H2_IN, (const float*)d_in[20], ws + WS_G6_GI + 3072,  2049 };
        b.j[3]  = { (const float*)d_in[19], st_h2,         (const float*)d_in[21], ws + WS_G6_GH + 3072,  1024 };
        // k_rnn1
        b.j[4]  = { (const float*)d_in[22], ws + WS_K1_IN, (const float*)d_in[24], ws + WS_G6_GI + 6144,  2049 };
        b.j[5]  = { (const float*)d_in[23], st_k1,         (const float*)d_in[25], ws + WS_G6_GH + 6144,  1024 };
        // k_rnn2
        b.j[6]  = { (const float*)d_in[26], ws + WS_K2_IN, (const float*)d_in[28], ws + WS_G6_GI + 9216,  2049 };
        b.j[7]  = { (const float*)d_in[27], st_k2,         (const float*)d_in[29], ws + WS_G6_GH + 9216,  1024 };
        // f_rnn1
        b.j[8]  = { (const float*)d_in[30], ws + WS_F1_IN, (const float*)d_in[32], ws + WS_G6_GI + 12288, 1025 };
        b.j[9]  = { (const float*)d_in[31], st_f1,         (const float*)d_in[33], ws + WS_G6_GH + 12288, 1024 };
        // f_rnn2
        b.j[10] = { (const float*)d_in[34], ws + WS_F2_IN, (const float*)d_in[36], ws + WS_G6_GI + 15360, 1025 };
        b.j[11] = { (const float*)d_in[35], st_f2,         (const float*)d_in[37], ws + WS_G6_GH + 15360, 1024 };
        gemv16_wmma_kernel<<<dim3(48, 12), 128, 0, stream>>>(b);
    }
    {
        Gru6States g = { { st_h1, st_h2, st_k1, st_k2, st_f1, st_f2 } };
        gru_gate6_kernel<<<dim3(4, 6), 256, 0, stream>>>(ws + WS_G6_GI, ws + WS_G6_GH, g, ws + WS_S6);
    }

    // phase 5: heads, output order [h_out1(h1), k_out1(k1), f_out1(f1), h_out2(h2), k_out2(k2), f_out2(f2)]
    {
        Heads h;
        h.W[0] = (const float*)d_in[38]; h.b[0] = (const float*)d_in[39]; h.s[0] = ws + WS_S6 + 0 * 1024; // h_out1 . h1_s
        h.W[1] = (const float*)d_in[42]; h.b[1] = (const float*)d_in[43]; h.s[1] = ws + WS_S6 + 2 * 1024; // k_out1 . k1_s
        h.W[2] = (const float*)d_in[46]; h.b[2] = (const float*)d_in[47]; h.s[2] = ws + WS_S6 + 4 * 1024; // f_out1 . f1_s
        h.W[3] = (const float*)d_in[40]; h.b[3] = (const float*)d_in[41]; h.s[3] = ws + WS_S6 + 1 * 1024; // h_out2 . h2_s
        h.W[4] = (const float*)d_in[44]; h.b[4] = (const float*)d_in[45]; h.s[4] = ws + WS_S6 + 3 * 1024; // k_out2 . k2_s
        h.W[5] = (const float*)d_in[48]; h.b[5] = (const float*)d_in[49]; h.s[5] = ws + WS_S6 + 5 * 1024; // f_out2 . f2_s
        out_heads_kernel<<<6, 256, 0, stream>>>(h, (float*)d_out);
    }
}